// DotProductAttention_23794118820323
// MI455X (gfx1250) — compile-verified
//
#include <hip/hip_runtime.h>
#include <math.h>

#define NN   100000
#define FF   64
#define HH   2
#define EE   1000000
#define COLS 256           // 2*H*F

typedef __attribute__((ext_vector_type(2))) float v2f;
typedef __attribute__((ext_vector_type(8))) float v8f;

// ---------------------------------------------------------------------------
// Kernel 1: qk[N][256] = x[N][64] @ W[64][256] + b   (fp32 WMMA 16x16x4)
// One wave per 16x16 output tile; K=64 consumed as 16 WMMAs of K=4.
// Grid is exact -> EXEC all-1s for every WMMA.
// ---------------------------------------------------------------------------
__global__ void __launch_bounds__(256)
gat_qk_gemm(const float* __restrict__ x, const float* __restrict__ W,
            const float* __restrict__ bias, float* __restrict__ qk) {
  const int wave = (blockIdx.x * blockDim.x + threadIdx.x) >> 5;
  const int lane = threadIdx.x & 31;
  const int nt   = wave & 15;        // 16 column tiles of 16
  const int mt   = wave >> 4;        // 6250 row tiles of 16
  const int half = lane >> 4;
  const int lo   = lane & 15;
  const int row0 = mt * 16;
  const int col0 = nt * 16;

  v8f acc = {};
  // A: lane holds x[row0+lo][k0 + 2*half + {0,1}]
  const float* xr = x + (size_t)(row0 + lo) * FF + 2 * half;
  // B: lane holds W[k0 + 2*half + {0,1}][col0+lo]
  const float* wc = W + (size_t)(2 * half) * COLS + col0 + lo;

#pragma unroll
  for (int k0 = 0; k0 < FF; k0 += 4) {
    v2f a, bm;
    a.x  = xr[k0 + 0];
    a.y  = xr[k0 + 1];
    bm.x = wc[(size_t)k0 * COLS];
    bm.y = wc[(size_t)(k0 + 1) * COLS];
    acc = __builtin_amdgcn_wmma_f32_16x16x4_f32(false, a, false, bm,
                                                (short)0, acc, false, false);
  }

  const float bv = bias[col0 + lo];
#pragma unroll
  for (int v = 0; v < 8; ++v) {
    const int r = row0 + v + 8 * half;          // D: VGPR v -> M = v + 8*half
    qk[(size_t)r * COLS + col0 + lo] = acc[v] + bv;
  }
}

// ---------------------------------------------------------------------------
// Kernel 0: init segment buffers (m = -inf, denom = 0)
// ---------------------------------------------------------------------------
__global__ void gat_init(float* __restrict__ m, float* __restrict__ denom) {
  int i = blockIdx.x * blockDim.x + threadIdx.x;
  if (i < NN * HH) { m[i] = -__builtin_inff(); denom[i] = 0.0f; }
}

// ---------------------------------------------------------------------------
// Kernel 2: per-edge scores + segment max.  One wave per edge:
//   lanes 0-15 = head 0, lanes 16-31 = head 1, each lane a float4 of F.
//   q[n,h,f] = qk[n][h*128 + f]; k[n,h,f] = qk[n][h*128 + 64 + f]
// ---------------------------------------------------------------------------
__global__ void __launch_bounds__(256)
gat_scores(const float* __restrict__ qk, const int* __restrict__ ei,
           float* __restrict__ sc, float* __restrict__ m) {
  const int e    = (blockIdx.x * blockDim.x + threadIdx.x) >> 5;
  const int lane = threadIdx.x & 31;
  const int h    = lane >> 4;
  const int f4   = (lane & 15) << 2;
  const int row  = ei[e];
  const int col  = ei[EE + e];

  const float4 q4 = *reinterpret_cast<const float4*>(
      qk + (size_t)row * COLS + h * 128 + f4);
  const float4 k4 = *reinterpret_cast<const float4*>(
      qk + (size_t)col * COLS + h * 128 + 64 + f4);

  float p = q4.x * k4.x + q4.y * k4.y + q4.z * k4.z + q4.w * k4.w;
  // reduce within each 16-lane half (masks < 16 keep lanes in-half)
  p += __shfl_xor(p, 8, 32);
  p += __shfl_xor(p, 4, 32);
  p += __shfl_xor(p, 2, 32);
  p += __shfl_xor(p, 1, 32);

  if ((lane & 15) == 0) {
    sc[(size_t)e * HH + h] = p;
    __hip_atomic_fetch_max(&m[row * HH + h], p, __ATOMIC_RELAXED,
                           __HIP_MEMORY_SCOPE_AGENT);
  }
}

// ---------------------------------------------------------------------------
// Kernel 3: m = isfinite(m) ? m : 0
// ---------------------------------------------------------------------------
__global__ void gat_fix_m(float* __restrict__ m) {
  int i = blockIdx.x * blockDim.x + threadIdx.x;
  if (i < NN * HH) {
    float v = m[i];
    m[i] = __builtin_isfinite(v) ? v : 0.0f;
  }
}

// ---------------------------------------------------------------------------
// Kernel 4: e = exp(score - m[row]) ; denom[row,h] += e ; sc overwritten by e
// One thread per (edge, head).
// ---------------------------------------------------------------------------
__global__ void gat_expsum(const int* __restrict__ ei, float* __restrict__ sc,
                           const float* __restrict__ m,
                           float* __restrict__ denom) {
  int idx = blockIdx.x * blockDim.x + threadIdx.x;   // 0 .. 2E-1
  if (idx < EE * HH) {
    int e = idx >> 1, h = idx & 1;
    int row = ei[e];
    float ev = __expf(sc[idx] - m[row * HH + h]);
    sc[idx] = ev;
    atomicAdd(&denom[row * HH + h], ev);
  }
}

// ---------------------------------------------------------------------------
// Kernel 5: out[e] = mean_h( e[e,h] / denom[row,h] )
// ---------------------------------------------------------------------------
__global__ void gat_out(const int* __restrict__ ei, const float* __restrict__ sc,
                        const float* __restrict__ denom, float* __restrict__ out) {
  int e = blockIdx.x * blockDim.x + threadIdx.x;
  if (e < EE) {
    int row = ei[e];
    float a0 = sc[(size_t)e * 2 + 0] / denom[row * 2 + 0];
    float a1 = sc[(size_t)e * 2 + 1] / denom[row * 2 + 1];
    out[e] = 0.5f * (a0 + a1);
  }
}

// ---------------------------------------------------------------------------
extern "C" void kernel_launch(void* const* d_in, const int* in_sizes, int n_in,
                              void* d_out, int out_size, void* d_ws,
                              size_t ws_size, hipStream_t stream) {
  const float* x  = (const float*)d_in[0];   // N*F
  const int*   ei = (const int*)d_in[1];     // 2*E
  const float* W  = (const float*)d_in[2];   // F * 256
  const float* b  = (const float*)d_in[3];   // 256
  float* out = (float*)d_out;                // E

  float* ws    = (float*)d_ws;
  float* qk    = ws;                                   // N*256 floats
  float* sc    = qk + (size_t)NN * COLS;               // E*H floats
  float* m     = sc + (size_t)EE * HH;                 // N*H floats
  float* denom = m + (size_t)NN * HH;                  // N*H floats

  // init segment buffers
  gat_init<<<(NN * HH + 255) / 256, 256, 0, stream>>>(m, denom);

  // projection GEMM: (N/16)*16 tiles -> 100000 waves -> 12500 blocks of 256
  gat_qk_gemm<<<(NN / 16) * 16 / 8, 256, 0, stream>>>(x, W, b, qk);

  // one wave per edge: E*32 threads / 256 = 125000 blocks (exact)
  gat_scores<<<EE / 8, 256, 0, stream>>>(qk, ei, sc, m);

  gat_fix_m<<<(NN * HH + 255) / 256, 256, 0, stream>>>(m);

  gat_expsum<<<(EE * HH + 255) / 256, 256, 0, stream>>>(ei, sc, m, denom);

  gat_out<<<(EE + 255) / 256, 256, 0, stream>>>(ei, sc, denom, out);
}